// BalancedSpatialGNN_17188459119259
// MI455X (gfx1250) — compile-verified
//
#include <hip/hip_runtime.h>
#include <hip/hip_bf16.h>

#define NODES 50000
#define EDGES 800000
#define EPSF  1e-5f

typedef __attribute__((ext_vector_type(2))) float v2f;
typedef __attribute__((ext_vector_type(8))) float v8f;

// ---------------- degree ----------------
__global__ void k_init_deg(float* deg, int n) {
    int i = blockIdx.x * blockDim.x + threadIdx.x;
    if (i < n) deg[i] = 1.0f;                       // self-loop
}
__global__ void k_deg_accum(const long long* ei, float* deg, int E) {
    int e = blockIdx.x * blockDim.x + threadIdx.x;
    if (e < E) atomicAdd(&deg[(int)ei[(size_t)E + e]], 1.0f);
}
__global__ void k_rsqrt(const float* deg, float* dis, int n) {
    int i = blockIdx.x * blockDim.x + threadIdx.x;
    if (i < n) dis[i] = rsqrtf(deg[i]);
}

// ---------------- layernorm over 256 feats, one wave per node ----------------
__global__ void k_ln256(const float* __restrict__ x, const float* __restrict__ g,
                        const float* __restrict__ b, float* __restrict__ h0, int n) {
    int wid  = (blockIdx.x * blockDim.x + threadIdx.x) >> 5;
    int lane = threadIdx.x & 31;
    if (wid >= n) return;
    const float* xr = x + (size_t)wid * 256;
    float v[8]; float s = 0.f;
    #pragma unroll
    for (int i = 0; i < 8; ++i) { v[i] = xr[lane + 32 * i]; s += v[i]; }
    #pragma unroll
    for (int o = 16; o > 0; o >>= 1) s += __shfl_xor(s, o, 32);
    float mu = s * (1.0f / 256.0f);
    float q = 0.f;
    #pragma unroll
    for (int i = 0; i < 8; ++i) { float d = v[i] - mu; q += d * d; }
    #pragma unroll
    for (int o = 16; o > 0; o >>= 1) q += __shfl_xor(q, o, 32);
    float inv = rsqrtf(q * (1.0f / 256.0f) + EPSF);
    float* hr = h0 + (size_t)wid * 256;
    #pragma unroll
    for (int i = 0; i < 8; ++i) {
        int f = lane + 32 * i;
        hr[f] = (v[i] - mu) * inv * g[f] + b[f];
    }
}

// ---------------- fp32 WMMA GEMM: out[M,N] = A[M,K] @ W[K,N] ----------------
// Compile-time K,N: constant strides, clean unrolled v_wmma chain.
// grid.x = M/16 (M multiple of 16); blockDim = 32 * NT, NT = ceil(N/16).
template <int K, int N, int NT>
__global__ void k_gemm_wmma_f32(const float* __restrict__ A, const float* __restrict__ W,
                                float* __restrict__ out) {
    __shared__ float Alds[16 * K];
    const int m0   = blockIdx.x * 16;
    const int tid  = threadIdx.x;
    const int lane = tid & 31;
    const int wave = tid >> 5;

    #pragma unroll 1
    for (int i = tid; i < 16 * K; i += 32 * NT) {
        int r = i / K, c = i - r * K;
        Alds[i] = A[(size_t)(m0 + r) * K + c];
    }
    __syncthreads();

    const int n0   = wave * 16;
    const int half = lane >> 4;                 // 0: K pair {k,k+1}; 1: {k+2,k+3}
    const int r    = lane & 15;
    const bool ok  = (n0 + r) < N;              // only layer-3 tile 1 is partial
    // clamp column so every lane loads in-bounds; mask after load (no EXEC divergence)
    const float* __restrict__ wcol = W + (ok ? (n0 + r) : (N - 1));
    const float* __restrict__ arow = Alds + r * K + 2 * half;

    v8f c = {};
    #pragma unroll 4
    for (int k0 = 0; k0 < K; k0 += 4) {
        const int ka = 2 * half;                // lane-half K offset within the 4-wide step
        v2f a, bfrag;
        a.x = arow[k0];
        a.y = arow[k0 + 1];
        float bx = wcol[(k0 + ka) * N];
        float by = wcol[(k0 + ka + 1) * N];
        bfrag.x = ok ? bx : 0.0f;
        bfrag.y = ok ? by : 0.0f;
        c = __builtin_amdgcn_wmma_f32_16x16x4_f32(false, a, false, bfrag,
                                                  (short)0, c, false, false);
    }
    if (ok) {
        #pragma unroll
        for (int j = 0; j < 8; ++j)
            out[(size_t)(m0 + j + 8 * half) * N + n0 + r] = c[j];
    }
}

// ---------------- agg = hW * (1/deg)  (self-loop init) ----------------
template <int F>
__global__ void k_selfinit(const float* __restrict__ hW, const float* __restrict__ dis,
                           float* __restrict__ agg, int n) {
    int i = blockIdx.x * blockDim.x + threadIdx.x;
    if (i >= n * F) return;
    int node = i / F;
    float d = dis[node];
    agg[i] = hW[i] * d * d;
}

// ---------------- edge scatter: agg[dst] += hW[src] * dis[src]*dis[dst] ----------------
template <int F>
__global__ void k_scatter(const float* __restrict__ hW, const long long* __restrict__ ei,
                          const float* __restrict__ dis, float* __restrict__ agg, int E) {
    int tid = blockIdx.x * blockDim.x + threadIdx.x;
    if (tid >= E * F) return;
    int e = tid / F, f = tid - e * F;
    int s = (int)ei[e];
    int d = (int)ei[(size_t)E + e];
    float nrm = dis[s] * dis[d];
    atomicAdd(&agg[(size_t)d * F + f], hW[(size_t)s * F + f] * nrm);
}

// ---------------- h = relu(BN(agg + bias)) ----------------
template <int F>
__global__ void k_bias_bn_relu(const float* __restrict__ agg, const float* __restrict__ bias,
                               const float* __restrict__ g, const float* __restrict__ bb,
                               const float* __restrict__ m, const float* __restrict__ v,
                               float* __restrict__ h, int n) {
    int i = blockIdx.x * blockDim.x + threadIdx.x;
    if (i >= n * F) return;
    int f = i % F;
    float t = (agg[i] + bias[f] - m[f]) * rsqrtf(v[f] + EPSF) * g[f] + bb[f];
    h[i] = t > 0.f ? t : 0.f;
}

// ---------------- classifier: fc1(24->12) + LN + relu + fc2(12->8) ----------------
__global__ void k_classifier(const float* __restrict__ h3,
                             const float* __restrict__ W1, const float* __restrict__ b1,
                             const float* __restrict__ lg, const float* __restrict__ lb,
                             const float* __restrict__ W2, const float* __restrict__ b2,
                             float* __restrict__ out, int n) {
    int node = blockIdx.x * blockDim.x + threadIdx.x;
    if (node >= n) return;
    float h[24];
    #pragma unroll
    for (int k = 0; k < 24; ++k) h[k] = h3[(size_t)node * 24 + k];
    float y[12];
    #pragma unroll
    for (int j = 0; j < 12; ++j) {
        float acc = b1[j];
        #pragma unroll
        for (int k = 0; k < 24; ++k) acc += h[k] * W1[k * 12 + j];
        y[j] = acc;
    }
    float mu = 0.f;
    #pragma unroll
    for (int j = 0; j < 12; ++j) mu += y[j];
    mu *= (1.0f / 12.0f);
    float q = 0.f;
    #pragma unroll
    for (int j = 0; j < 12; ++j) { float d = y[j] - mu; q += d * d; }
    float inv = rsqrtf(q * (1.0f / 12.0f) + EPSF);
    float t[12];
    #pragma unroll
    for (int j = 0; j < 12; ++j) {
        float z = (y[j] - mu) * inv * lg[j] + lb[j];
        t[j] = z > 0.f ? z : 0.f;
    }
    #pragma unroll
    for (int c2 = 0; c2 < 8; ++c2) {
        float acc = b2[c2];
        #pragma unroll
        for (int j = 0; j < 12; ++j) acc += t[j] * W2[j * 8 + c2];
        out[(size_t)node * 8 + c2] = acc;
    }
}

static inline int cdiv(long long a, int b) { return (int)((a + b - 1) / b); }

extern "C" void kernel_launch(void* const* d_in, const int* in_sizes, int n_in,
                              void* d_out, int out_size, void* d_ws, size_t ws_size,
                              hipStream_t stream) {
    const float*     x    = (const float*)d_in[0];
    const long long* ei   = (const long long*)d_in[1];   // [2,E] int64
    const float*     ln_g = (const float*)d_in[2];
    const float*     ln_b = (const float*)d_in[3];
    const float*     W1   = (const float*)d_in[4];
    const float*     b1   = (const float*)d_in[5];
    const float*     bn1g = (const float*)d_in[6];
    const float*     bn1b = (const float*)d_in[7];
    const float*     bn1m = (const float*)d_in[8];
    const float*     bn1v = (const float*)d_in[9];
    const float*     W2   = (const float*)d_in[10];
    const float*     b2   = (const float*)d_in[11];
    const float*     bn2g = (const float*)d_in[12];
    const float*     bn2b = (const float*)d_in[13];
    const float*     bn2m = (const float*)d_in[14];
    const float*     bn2v = (const float*)d_in[15];
    const float*     W3   = (const float*)d_in[16];
    const float*     b3   = (const float*)d_in[17];
    const float*     bn3g = (const float*)d_in[18];
    const float*     bn3b = (const float*)d_in[19];
    const float*     bn3m = (const float*)d_in[20];
    const float*     bn3v = (const float*)d_in[21];
    const float*     fc1W = (const float*)d_in[22];
    const float*     fc1b = (const float*)d_in[23];
    const float*     lncg = (const float*)d_in[24];
    const float*     lncb = (const float*)d_in[25];
    const float*     fc2W = (const float*)d_in[26];
    const float*     fc2b = (const float*)d_in[27];
    float* out = (float*)d_out;

    const int Nn = NODES, E = EDGES;

    // workspace carve-up (all fp32): deg | dis | h0[N*256] | hW[N*96] | agg[N*96] | hcur[N*96]
    float* deg  = (float*)d_ws;
    float* dis  = deg  + Nn;
    float* h0   = dis  + Nn;
    float* hW   = h0   + (size_t)Nn * 256;
    float* agg  = hW   + (size_t)Nn * 96;
    float* hcur = agg  + (size_t)Nn * 96;

    // ---- degree + rsqrt ----
    k_init_deg <<<cdiv(Nn, 256), 256, 0, stream>>>(deg, Nn);
    k_deg_accum<<<cdiv(E, 256), 256, 0, stream>>>(ei, deg, E);
    k_rsqrt    <<<cdiv(Nn, 256), 256, 0, stream>>>(deg, dis, Nn);

    // ---- layernorm (one wave per node) ----
    k_ln256<<<cdiv(Nn, 8), 256, 0, stream>>>(x, ln_g, ln_b, h0, Nn);

    const int mtiles = Nn / 16;   // 3125, exact

    // ---- layer 1: K=256, F=96 ----
    k_gemm_wmma_f32<256, 96, 6><<<mtiles, 32 * 6, 0, stream>>>(h0, W1, hW);
    k_selfinit<96>     <<<cdiv((long long)Nn * 96, 256), 256, 0, stream>>>(hW, dis, agg, Nn);
    k_scatter<96>      <<<cdiv((long long)E * 96, 256), 256, 0, stream>>>(hW, ei, dis, agg, E);
    k_bias_bn_relu<96> <<<cdiv((long long)Nn * 96, 256), 256, 0, stream>>>(agg, b1, bn1g, bn1b, bn1m, bn1v, hcur, Nn);

    // ---- layer 2: K=96, F=48 ----
    k_gemm_wmma_f32<96, 48, 3><<<mtiles, 32 * 3, 0, stream>>>(hcur, W2, hW);
    k_selfinit<48>     <<<cdiv((long long)Nn * 48, 256), 256, 0, stream>>>(hW, dis, agg, Nn);
    k_scatter<48>      <<<cdiv((long long)E * 48, 256), 256, 0, stream>>>(hW, ei, dis, agg, E);
    k_bias_bn_relu<48> <<<cdiv((long long)Nn * 48, 256), 256, 0, stream>>>(agg, b2, bn2g, bn2b, bn2m, bn2v, hcur, Nn);

    // ---- layer 3: K=48, F=24 (second N-tile partially masked) ----
    k_gemm_wmma_f32<48, 24, 2><<<mtiles, 32 * 2, 0, stream>>>(hcur, W3, hW);
    k_selfinit<24>     <<<cdiv((long long)Nn * 24, 256), 256, 0, stream>>>(hW, dis, agg, Nn);
    k_scatter<24>      <<<cdiv((long long)E * 24, 256), 256, 0, stream>>>(hW, ei, dis, agg, E);
    k_bias_bn_relu<24> <<<cdiv((long long)Nn * 24, 256), 256, 0, stream>>>(agg, b3, bn3g, bn3b, bn3m, bn3v, hcur, Nn);

    // ---- classifier head ----
    k_classifier<<<cdiv(Nn, 256), 256, 0, stream>>>(hcur, fc1W, fc1b, lncg, lncb, fc2W, fc2b, out, Nn);
}